// LRURecurrentBlock_71305047048639
// MI455X (gfx1250) — compile-verified
//
#include <hip/hip_runtime.h>
#include <hip/hip_bf16.h>

// RG-LRU block for MI455X (gfx1250, wave32).
// GEMMs: V_WMMA_F32_16X16X4_F32 with 2x2 register blocking (32x32 per wave).
// A tiles staged in LDS via GLOBAL_LOAD_ASYNC_TO_LDS_B128 (ASYNCcnt path).
// Fragment layouts:
//   A (16x4 f32): lanes 0-15 {K0,K1} @ M=lane ; lanes 16-31 {K2,K3} @ M=lane-16
//   B (4x16 f32): lanes 0-15 {K0,K1} @ N=lane ; lanes 16-31 {K2,K3} @ N=lane-16
//   C/D (16x16 f32, 8 VGPRs): vgpr r -> M = r + 8*(lane>>4), N = lane&15

#define BT   (32 * 2048)
#define TT   2048
#define BB   32
#define WDIM 512
#define LRUW 512
#define HH   8
#define TWIN 4
#define KC   128          // K chunk staged in LDS

typedef float v2f __attribute__((ext_vector_type(2)));
typedef float v8f __attribute__((ext_vector_type(8)));

__device__ __forceinline__ float gelu_tanh(float x) {
    const float k0 = 0.7978845608028654f;   // sqrt(2/pi)
    const float k1 = 0.044715f;
    float inner = k0 * (x + k1 * x * x * x);
    return 0.5f * x * (1.0f + tanhf(inner));
}

__device__ __forceinline__ float sigmoidf(float x) {
    return 1.0f / (1.0f + __expf(-x));
}

// Async copy of 16B global -> LDS (ASYNCcnt-tracked, no VGPR round trip).
__device__ __forceinline__ void async_g2l_b128(void* lds_ptr, const float* gptr) {
    unsigned int       ldsoff = (unsigned int)(unsigned long long)lds_ptr;
    unsigned long long gaddr  = (unsigned long long)gptr;
    asm volatile("global_load_async_to_lds_b128 %0, %1, off"
                 :: "v"(ldsoff), "v"(gaddr) : "memory");
}
__device__ __forceinline__ void wait_async0() {
    asm volatile("s_wait_asynccnt 0x0" ::: "memory");
}

// ---------------------------------------------------------------------------
// Kernel 1: xb = x @ w_x + b_x ; yb = gelu(x @ w_y + b_y)
// Block = 128 threads (4 waves). Block tile: M=32, N=128 of the concatenated
// [w_x | w_y] N=1024 space. Each wave owns a 32x32 tile (2x2 WMMA subtiles).
// K staged in LDS in chunks of 128 via async loads.
// ---------------------------------------------------------------------------
__global__ __launch_bounds__(128)
void k_proj(const float* __restrict__ x,
            const float* __restrict__ w_x, const float* __restrict__ b_x,
            const float* __restrict__ w_y, const float* __restrict__ b_y,
            float* __restrict__ xb, float* __restrict__ yb)
{
    __shared__ float As[32][KC + 4];         // stride 132: conflict-free
    const int m0   = blockIdx.x * 32;
    const int tid  = threadIdx.x;
    const int wave = tid >> 5;
    const int lane = tid & 31;
    const int half = lane >> 4;
    const int lcol = lane & 15;

    const int n0 = blockIdx.y * 128 + wave * 32;         // 0..1023, 32-aligned
    const float* Wm = (n0 < WDIM) ? w_x : w_y;
    const int nc    = (n0 < WDIM) ? n0 : (n0 - WDIM);

    v8f acc00 = {}, acc01 = {}, acc10 = {}, acc11 = {};

    for (int kc = 0; kc < WDIM; kc += KC) {
        // async-stage the 32 x 128 A chunk into LDS (8 x b128 per thread)
        #pragma unroll
        for (int i = 0; i < 8; ++i) {
            int idx = tid + i * 128;         // 0..1023 float4 slots
            int row = idx >> 5;              // 32 float4 per row
            int c4  = idx & 31;
            async_g2l_b128(&As[row][c4 * 4],
                           x + (size_t)(m0 + row) * WDIM + kc + c4 * 4);
        }
        wait_async0();
        __syncthreads();

        #pragma unroll 4
        for (int kk = 0; kk < KC; kk += 4) {
            v2f a0, a1, b0, b1;
            a0.x = As[lcol][kk + 2 * half + 0];
            a0.y = As[lcol][kk + 2 * half + 1];
            a1.x = As[16 + lcol][kk + 2 * half + 0];
            a1.y = As[16 + lcol][kk + 2 * half + 1];
            const float* Wp = Wm + (size_t)(kc + kk + 2 * half) * LRUW + nc;
            b0.x = Wp[lcol];
            b0.y = Wp[LRUW + lcol];
            b1.x = Wp[16 + lcol];
            b1.y = Wp[LRUW + 16 + lcol];
            acc00 = __builtin_amdgcn_wmma_f32_16x16x4_f32(false, a0, false, b0,
                                                          (short)0, acc00, false, false);
            acc01 = __builtin_amdgcn_wmma_f32_16x16x4_f32(false, a0, false, b1,
                                                          (short)0, acc01, false, false);
            acc10 = __builtin_amdgcn_wmma_f32_16x16x4_f32(false, a1, false, b0,
                                                          (short)0, acc10, false, false);
            acc11 = __builtin_amdgcn_wmma_f32_16x16x4_f32(false, a1, false, b1,
                                                          (short)0, acc11, false, false);
        }
        __syncthreads();
    }

    if (n0 < WDIM) {
        float bia0 = b_x[nc + lcol];
        float bia1 = b_x[nc + 16 + lcol];
        #pragma unroll
        for (int r = 0; r < 8; ++r) {
            int m = m0 + r + 8 * half;
            xb[(size_t)m * LRUW + nc + lcol]             = acc00[r] + bia0;
            xb[(size_t)m * LRUW + nc + 16 + lcol]        = acc01[r] + bia1;
            xb[(size_t)(m + 16) * LRUW + nc + lcol]      = acc10[r] + bia0;
            xb[(size_t)(m + 16) * LRUW + nc + 16 + lcol] = acc11[r] + bia1;
        }
    } else {
        float bia0 = b_y[nc + lcol];
        float bia1 = b_y[nc + 16 + lcol];
        #pragma unroll
        for (int r = 0; r < 8; ++r) {
            int m = m0 + r + 8 * half;
            yb[(size_t)m * LRUW + nc + lcol]             = gelu_tanh(acc00[r] + bia0);
            yb[(size_t)m * LRUW + nc + 16 + lcol]        = gelu_tanh(acc01[r] + bia1);
            yb[(size_t)(m + 16) * LRUW + nc + lcol]      = gelu_tanh(acc10[r] + bia0);
            yb[(size_t)(m + 16) * LRUW + nc + 16 + lcol] = gelu_tanh(acc11[r] + bia1);
        }
    }
}

// ---------------------------------------------------------------------------
// Kernel 2: causal conv (TW=4) + block-diagonal gate GEMMs (wi, wa, 64x64 per
// head, via WMMA) + pointwise RG-LRU math. Grid = (BT/16, H). 256 threads.
// ---------------------------------------------------------------------------
__global__ __launch_bounds__(256)
void k_gates(const float* __restrict__ xb,
             const float* __restrict__ conv_w, const float* __restrict__ conv_b,
             const unsigned char* __restrict__ reset,
             const float* __restrict__ a_param,
             const float* __restrict__ wi, const float* __restrict__ bi,
             const float* __restrict__ wa, const float* __restrict__ ba,
             float* __restrict__ a_out, float* __restrict__ n_out)
{
    __shared__ float sX[16][68];     // conv output tile (16 rows x 64 channels)
    __shared__ float sG[16][130];    // gates: gx in [0..63], ga in [65..128]
    const int m0   = blockIdx.x * 16;
    const int head = blockIdx.y;
    const int tid  = threadIdx.x;
    const int wave = tid >> 5;       // 0..7
    const int lane = tid & 31;
    const int half = lane >> 4;
    const int lcol = lane & 15;
    const int bidx  = m0 / TT;       // tile never straddles a batch boundary
    const int tbase = m0 - bidx * TT;

    // causal conv into LDS
    #pragma unroll
    for (int e = 0; e < 4; ++e) {
        int idx = tid + e * 256;     // 0..1023
        int row = idx >> 6, col = idx & 63;
        int c = head * 64 + col;
        int t = tbase + row;
        float accv = conv_b[c];
        #pragma unroll
        for (int s = 0; s < TWIN; ++s) {
            int tt = t - (TWIN - 1) + s;
            if (tt >= 0)
                accv += conv_w[s * LRUW + c] *
                        xb[((size_t)bidx * TT + tt) * LRUW + c];
        }
        sX[row][col] = accv;
    }
    __syncthreads();

    // block-diagonal gate GEMMs: waves 0-3 -> wi (gate_x), waves 4-7 -> wa (gate_a)
    const float* Wm = ((wave < 4) ? wi : wa) + (size_t)head * 64 * 64;
    const int j0  = (wave & 3) * 16;
    const int dst = (wave < 4) ? 0 : 65;
    v8f acc = {};
    #pragma unroll
    for (int k0 = 0; k0 < 64; k0 += 4) {
        v2f a, b;
        a.x = sX[lcol][k0 + 2 * half + 0];
        a.y = sX[lcol][k0 + 2 * half + 1];
        b.x = Wm[(k0 + 2 * half + 0) * 64 + j0 + lcol];
        b.y = Wm[(k0 + 2 * half + 1) * 64 + j0 + lcol];
        acc = __builtin_amdgcn_wmma_f32_16x16x4_f32(false, a, false, b,
                                                    (short)0, acc, false, false);
    }
    const float* bias = ((wave < 4) ? bi : ba) + head * 64;
    float bj = bias[j0 + lcol];
    #pragma unroll
    for (int r = 0; r < 8; ++r)
        sG[r + 8 * half][dst + j0 + lcol] = sigmoidf(acc[r] + bj);
    __syncthreads();

    // pointwise RG-LRU math
    #pragma unroll
    for (int e = 0; e < 4; ++e) {
        int idx = tid + e * 256;
        int row = idx >> 6, col = idx & 63;
        int c = head * 64 + col;
        int m = m0 + row;
        float gx = sG[row][col];
        float ga = sG[row][65 + col];
        float sp    = log1pf(__expf(a_param[c]));        // softplus
        float log_a = -8.0f * ga * sp;
        float av    = __expf(log_a);
        float mult  = sqrtf(fmaxf(1.0f - __expf(2.0f * log_a), 0.0f));
        if (reset[m] != 0) { mult = 1.0f; av = 0.0f; }
        a_out[(size_t)m * LRUW + c] = av;
        n_out[(size_t)m * LRUW + c] = sX[row][col] * gx * mult;
    }
}

// ---------------------------------------------------------------------------
// Kernel 3: sequential scan h = a*h + normed over T, one thread per (b,c).
// [BT,512] layout keeps every per-step wave access fully coalesced (128B).
// Emits z = y * yb for the output GEMM.
// ---------------------------------------------------------------------------
__global__ __launch_bounds__(256)
void k_scan(const float* __restrict__ a, const float* __restrict__ nrm,
            const float* __restrict__ yb, const float* __restrict__ h0,
            float* __restrict__ z)
{
    int g = blockIdx.x * 256 + threadIdx.x;   // 0 .. B*512-1
    int b = g >> 9;
    int c = g & 511;
    float h = h0[b * LRUW + c];
    size_t base = ((size_t)b * TT) * LRUW + c;
    for (int t = 0; t < TT; ++t) {
        size_t idx = base + (size_t)t * LRUW;
        h = fmaf(a[idx], h, nrm[idx]);
        z[idx] = h * yb[idx];
    }
}

// ---------------------------------------------------------------------------
// Kernel 4: out = z @ w_out + b_out  (same tiling as k_proj, N=512)
// ---------------------------------------------------------------------------
__global__ __launch_bounds__(128)
void k_out(const float* __restrict__ z,
           const float* __restrict__ w_out, const float* __restrict__ b_out,
           float* __restrict__ out)
{
    __shared__ float As[32][KC + 4];
    const int m0   = blockIdx.x * 32;
    const int tid  = threadIdx.x;
    const int wave = tid >> 5;
    const int lane = tid & 31;
    const int half = lane >> 4;
    const int lcol = lane & 15;

    const int n0 = blockIdx.y * 128 + wave * 32;   // 0..511

    v8f acc00 = {}, acc01 = {}, acc10 = {}, acc11 = {};

    for (int kc = 0; kc < LRUW; kc += KC) {
        #pragma unroll
        for (int i = 0; i < 8; ++i) {
            int idx = tid + i * 128;
            int row = idx >> 5;
            int c4  = idx & 31;
            async_g2l_b128(&As[row][c4 * 4],
                           z + (size_t)(m0 + row) * LRUW + kc + c4 * 4);
        }
        wait_async0();
        __syncthreads();

        #pragma unroll 4
        for (int kk = 0; kk < KC; kk += 4) {
            v2f a0, a1, b0, b1;
            a0.x = As[lcol][kk + 2 * half + 0];
            a0.y = As[lcol][kk + 2 * half + 1];
            a1.x = As[16 + lcol][kk + 2 * half + 0];
            a1.y = As[16 + lcol][kk + 2 * half + 1];
            const float* Wp = w_out + (size_t)(kc + kk + 2 * half) * WDIM + n0;
            b0.x = Wp[lcol];
            b0.y = Wp[WDIM + lcol];
            b1.x = Wp[16 + lcol];
            b1.y = Wp[WDIM + 16 + lcol];
            acc00 = __builtin_amdgcn_wmma_f32_16x16x4_f32(false, a0, false, b0,
                                                          (short)0, acc00, false, false);
            acc01 = __builtin_amdgcn_wmma_f32_16x16x4_f32(false, a0, false, b1,
                                                          (short)0, acc01, false, false);
            acc10 = __builtin_amdgcn_wmma_f32_16x16x4_f32(false, a1, false, b0,
                                                          (short)0, acc10, false, false);
            acc11 = __builtin_amdgcn_wmma_f32_16x16x4_f32(false, a1, false, b1,
                                                          (short)0, acc11, false, false);
        }
        __syncthreads();
    }

    float bia0 = b_out[n0 + lcol];
    float bia1 = b_out[n0 + 16 + lcol];
    #pragma unroll
    for (int r = 0; r < 8; ++r) {
        int m = m0 + r + 8 * half;
        out[(size_t)m * WDIM + n0 + lcol]             = acc00[r] + bia0;
        out[(size_t)m * WDIM + n0 + 16 + lcol]        = acc01[r] + bia1;
        out[(size_t)(m + 16) * WDIM + n0 + lcol]      = acc10[r] + bia0;
        out[(size_t)(m + 16) * WDIM + n0 + 16 + lcol] = acc11[r] + bia1;
    }
}

// ---------------------------------------------------------------------------
extern "C" void kernel_launch(void* const* d_in, const int* in_sizes, int n_in,
                              void* d_out, int out_size, void* d_ws, size_t ws_size,
                              hipStream_t stream)
{
    const float*         x       = (const float*)d_in[0];
    const unsigned char* reset   = (const unsigned char*)d_in[1];
    const float*         h0      = (const float*)d_in[2];
    const float*         w_x     = (const float*)d_in[3];
    const float*         b_x     = (const float*)d_in[4];
    const float*         w_y     = (const float*)d_in[5];
    const float*         b_y     = (const float*)d_in[6];
    const float*         conv_w  = (const float*)d_in[7];
    const float*         conv_b  = (const float*)d_in[8];
    const float*         a_param = (const float*)d_in[9];
    const float*         wi      = (const float*)d_in[10];
    const float*         bi      = (const float*)d_in[11];
    const float*         wa      = (const float*)d_in[12];
    const float*         ba      = (const float*)d_in[13];
    const float*         w_out   = (const float*)d_in[14];
    const float*         b_out   = (const float*)d_in[15];

    const size_t NE = (size_t)BT * LRUW;
    float* xb = (float*)d_ws;       // [BT,512]
    float* yb = xb + NE;            // [BT,512]
    float* aa = yb + NE;            // [BT,512]
    float* nn = aa + NE;            // [BT,512]
    float* zz = xb;                 // reuse xb after k_gates consumed it

    dim3 g1(BT / 32, 1024 / 128);
    k_proj<<<g1, 128, 0, stream>>>(x, w_x, b_x, w_y, b_y, xb, yb);

    dim3 g2(BT / 16, HH);
    k_gates<<<g2, 256, 0, stream>>>(xb, conv_w, conv_b, reset, a_param,
                                    wi, bi, wa, ba, aa, nn);

    k_scan<<<(BB * LRUW) / 256, 256, 0, stream>>>(aa, nn, yb, h0, zz);

    dim3 g4(BT / 32, WDIM / 128);
    k_out<<<g4, 128, 0, stream>>>(zz, w_out, b_out, (float*)d_out);
}